// InterpretableMultiHeadAttention_4389456577130
// MI455X (gfx1250) — compile-verified
//
#include <hip/hip_runtime.h>

typedef __attribute__((ext_vector_type(2))) float v2f;
typedef __attribute__((ext_vector_type(8))) float v8f;

#define NHEADS 16
#define DH     64
#define SEQ    1024
#define DMODEL 1024
#define BATCH  4

__device__ __forceinline__ v8f wmma_f32(v2f a, v2f b, v8f c) {
  // V_WMMA_F32_16X16X4_F32: D = A(16x4) * B(4x16) + C(16x16), fp32 exact
  return __builtin_amdgcn_wmma_f32_16x16x4_f32(false, a, false, b, (short)0, c,
                                               false, false);
}

// ---------------------------------------------------------------------------
// Generic GEMM + bias: C[M,N] = A[M,K] @ W[K,N] + bias[N]
// One 16x16 C tile per wave, 8 waves / block. fp32 WMMA, K stepped by 4.
// ---------------------------------------------------------------------------
__global__ __launch_bounds__(256) void gemm_bias_kernel(
    const float* __restrict__ A, const float* __restrict__ W,
    const float* __restrict__ bias, float* __restrict__ C, int M, int K, int N) {
  int tid  = threadIdx.x;
  int wave = tid >> 5;
  int lane = tid & 31;
  int lo = lane & 15, hi = lane >> 4;

  int tilesN = N >> 4;
  int tile = blockIdx.x * 8 + wave;
  int tm = tile / tilesN, tn = tile - tm * tilesN;
  int m0 = tm << 4, n0 = tn << 4;
  if (m0 >= M) return;

  const float* arow = A + (size_t)(m0 + lo) * K;   // A-frag row for this lane
  const float* bcol = W + n0 + lo;                 // B-frag col for this lane

  v8f acc = {};
  for (int kk = 0; kk < K; kk += 4) {
    int kb = kk + 2 * hi;                          // lanes 16-31 carry K+2,K+3
    v2f a = *(const v2f*)(arow + kb);              // A[m, kb], A[m, kb+1]
    v2f b;
    b.x = bcol[(size_t)kb * N];                    // W[kb,   n]
    b.y = bcol[(size_t)(kb + 1) * N];              // W[kb+1, n]
    acc = wmma_f32(a, b, acc);
  }
  float bv = bias[n0 + lo];
#pragma unroll
  for (int i = 0; i < 8; ++i) {
    int row = m0 + i + 8 * hi;                     // C layout: vgpr i, half hi
    C[(size_t)row * N + n0 + lo] = acc[i] + bv;
  }
}

// ---------------------------------------------------------------------------
// v_shared[b,s,d] = mean_h V[b,s,h*64+d]
// ---------------------------------------------------------------------------
__global__ void vshared_kernel(const float* __restrict__ V,
                               float* __restrict__ Vs) {
  int idx = blockIdx.x * blockDim.x + threadIdx.x;  // over B*S*DH
  if (idx >= BATCH * SEQ * DH) return;
  int d  = idx & (DH - 1);
  int bs = idx >> 6;
  const float* row = V + (size_t)bs * DMODEL;
  float s = 0.f;
#pragma unroll
  for (int h = 0; h < NHEADS; ++h) s += row[h * DH + d];
  Vs[idx] = s * (1.0f / NHEADS);
}

__global__ void zero_kernel(float* __restrict__ p, int n) {
  int i = blockIdx.x * blockDim.x + threadIdx.x;
  if (i < n) p[i] = 0.f;
}

// ---------------------------------------------------------------------------
// Fused attention for one (b, h, 16-row q tile):
//  logits (WMMA) -> sparsemax (bitonic sort + scan in LDS) -> avg_attn atomics
//  -> probs @ v_shared (WMMA) -> OutH
// ---------------------------------------------------------------------------
__global__ __launch_bounds__(256) void attn_kernel(
    const float* __restrict__ Q, const float* __restrict__ K,
    const float* __restrict__ Vs, float* __restrict__ OutH,
    float* __restrict__ Avg) {
  __shared__ float LG[16][SEQ + 1];     // logits -> probs (padded vs bank conflicts)
  __shared__ float SORT[16][SEQ + 1];   // sorted copy -> cumsum
  __shared__ float csum[16][16];
  __shared__ int   cnt[16][16];
  __shared__ float tauv[16];

  int blk = blockIdx.x;                 // b * (H * S/16) + h * (S/16) + qt
  int qt = blk & 63;
  int h  = (blk >> 6) & 15;
  int b  = blk >> 10;
  int q0 = qt << 4;

  int tid  = threadIdx.x;
  int wave = tid >> 5;
  int lane = tid & 31;
  int lo = lane & 15, hi = lane >> 4;

  // ---- Phase 1: logits tile (16 x 1024) = Qt(16x64) @ Kt^T, scaled 1/8 ----
  const float* qbase = Q + (size_t)(b * SEQ + q0 + lo) * DMODEL + h * DH;
  v2f afr[16];
#pragma unroll
  for (int ds = 0; ds < 16; ++ds) {
    int kb = ds * 4 + 2 * hi;
    afr[ds] = *(const v2f*)(qbase + kb);
  }
  for (int nt = 0; nt < 8; ++nt) {      // each wave: 8 N-tiles = 128 columns
    int n0 = (wave * 8 + nt) << 4;
    const float* kp = K + (size_t)(b * SEQ + n0 + lo) * DMODEL + h * DH;
    v8f acc = {};
#pragma unroll
    for (int ds = 0; ds < 16; ++ds) {
      int kb = ds * 4 + 2 * hi;
      v2f bf = *(const v2f*)(kp + kb);
      acc = wmma_f32(afr[ds], bf, acc);
    }
#pragma unroll
    for (int i = 0; i < 8; ++i)
      LG[i + 8 * hi][n0 + lo] = acc[i] * 0.125f;   // 1/sqrt(64)
  }
  __syncthreads();

  // ---- Phase 2a: copy for sorting ----
  for (int t = tid; t < 16 * SEQ; t += 256) {
    int r = t >> 10, c = t & 1023;
    SORT[r][c] = LG[r][c];
  }
  __syncthreads();

  // ---- Phase 2b: bitonic sort, descending, 16 rows in parallel ----
  for (int ksz = 2; ksz <= SEQ; ksz <<= 1) {
    for (int j = ksz >> 1; j > 0; j >>= 1) {
      for (int t = tid; t < 16 * 512; t += 256) {
        int r = t >> 9;
        int p = t & 511;
        int low = p & (j - 1);
        int idx = ((p - low) << 1) + low;
        int par = idx + j;
        float x0 = SORT[r][idx];
        float x1 = SORT[r][par];
        bool descBlk = ((idx & ksz) == 0);
        if (descBlk ? (x0 < x1) : (x0 > x1)) {
          SORT[r][idx] = x1;
          SORT[r][par] = x0;
        }
      }
      __syncthreads();
    }
  }

  // ---- Phase 2c: per-row cumsum, support size k_z, tau ----
  {
    int r = tid >> 4, c = tid & 15;
    int base = c * 64;
    float s = 0.f;
    for (int e = 0; e < 64; ++e) s += SORT[r][base + e];
    csum[r][c] = s;
  }
  __syncthreads();
  {
    int r = tid >> 4, c = tid & 15;
    float run = 0.f;
    for (int i = 0; i < c; ++i) run += csum[r][i];
    int base = c * 64;
    int count = 0;
    for (int e = 0; e < 64; ++e) {
      int i = base + e;
      float z = SORT[r][i];
      run += z;
      if (1.0f + (float)(i + 1) * z > run) ++count;  // support condition
      SORT[r][i] = run;                              // overwrite with cumsum
    }
    cnt[r][c] = count;
  }
  __syncthreads();
  if (tid < 16) {
    int r = tid;
    int kz = 0;
    for (int i = 0; i < 16; ++i) kz += cnt[r][i];
    float csk = SORT[r][kz - 1];
    tauv[r] = (csk - 1.0f) / (float)kz;
  }
  __syncthreads();

  // ---- Phase 3: probs = max(z - tau, 0); accumulate avg_attention ----
  {
    int r = tid >> 4, c = tid & 15;
    float tau = tauv[r];
    float* avgrow = Avg + (size_t)(b * SEQ + q0 + r) * SEQ;
    int base = c * 64;
    for (int e = 0; e < 64; ++e) {
      int i = base + e;
      float p = LG[r][i] - tau;
      p = p > 0.f ? p : 0.f;
      LG[r][i] = p;
      float contrib = p * (1.0f / NHEADS);
      float* addr = avgrow + i;
      asm volatile("global_atomic_add_f32 %0, %1, off"
                   :
                   : "v"(addr), "v"(contrib)
                   : "memory");
    }
  }
  __syncthreads();

  // ---- Phase 4: out_h tile = probs(16x1024) @ Vs(1024x64) ----
  if (wave < 4) {
    int n0 = wave << 4;
    const float* vsb = Vs + (size_t)b * SEQ * DH + n0 + lo;
    v8f acc = {};
    for (int kk = 0; kk < SEQ; kk += 4) {
      int kb = kk + 2 * hi;
      v2f a;
      a.x = LG[lo][kb];
      a.y = LG[lo][kb + 1];
      v2f bf;
      bf.x = vsb[(size_t)kb * DH];
      bf.y = vsb[(size_t)(kb + 1) * DH];
      acc = wmma_f32(a, bf, acc);
    }
#pragma unroll
    for (int i = 0; i < 8; ++i)
      OutH[(size_t)(b * SEQ + q0 + i + 8 * hi) * DMODEL + h * DH + n0 + lo] =
          acc[i];
  }
}

// ---------------------------------------------------------------------------
extern "C" void kernel_launch(void* const* d_in, const int* in_sizes, int n_in,
                              void* d_out, int out_size, void* d_ws,
                              size_t ws_size, hipStream_t stream) {
  const float* x  = (const float*)d_in[0];
  const float* Wq = (const float*)d_in[1];
  const float* bq = (const float*)d_in[2];
  const float* Wk = (const float*)d_in[3];
  const float* bk = (const float*)d_in[4];
  const float* Wv = (const float*)d_in[5];
  const float* bv = (const float*)d_in[6];
  const float* Wo = (const float*)d_in[7];
  const float* bo = (const float*)d_in[8];

  float* out  = (float*)d_out;
  float* xout = out;                                    // (B,S,D)
  float* avg  = out + (size_t)BATCH * SEQ * DMODEL;     // (B,S,S)

  const size_t MD = (size_t)BATCH * SEQ * DMODEL;       // 4M floats
  float* Qb   = (float*)d_ws;
  float* Kb   = Qb + MD;
  float* Vb   = Kb + MD;
  float* Vs   = Vb + MD;                                // B*S*DH
  float* OutH = Vs + (size_t)BATCH * SEQ * DH;

  const int M = BATCH * SEQ;                            // 4096
  const int tiles = (M / 16) * (DMODEL / 16);           // 16384
  dim3 blk(256);
  dim3 gGemm(tiles / 8);                                // 2048 blocks

  gemm_bias_kernel<<<gGemm, blk, 0, stream>>>(x, Wq, bq, Qb, M, DMODEL, DMODEL);
  gemm_bias_kernel<<<gGemm, blk, 0, stream>>>(x, Wk, bk, Kb, M, DMODEL, DMODEL);
  gemm_bias_kernel<<<gGemm, blk, 0, stream>>>(x, Wv, bv, Vb, M, DMODEL, DMODEL);

  vshared_kernel<<<(BATCH * SEQ * DH) / 256, blk, 0, stream>>>(Vb, Vs);

  zero_kernel<<<(BATCH * SEQ * SEQ) / 256, blk, 0, stream>>>(
      avg, BATCH * SEQ * SEQ);

  attn_kernel<<<BATCH * NHEADS * (SEQ / 16), blk, 0, stream>>>(Qb, Kb, Vs,
                                                               OutH, avg);

  gemm_bias_kernel<<<gGemm, blk, 0, stream>>>(OutH, Wo, bo, xout, M, DMODEL,
                                              DMODEL);
}